// ProceduralMemory_3075196584326
// MI455X (gfx1250) — compile-verified
//
#include <hip/hip_runtime.h>
#include <math.h>

#define BS 64
#define RREG 512
#define DH 1024
#define RHO 0.95f
#define DECAYC 0.99f
#define A_MAXC 2.0f
#define BUDGETC 64.0f
#define TOPKC 8
#define TAUC 0.1f
#define WEAKW 0.1f
#define EPSV 1e-8f

typedef float v2f __attribute__((ext_vector_type(2)));
typedef float v8f __attribute__((ext_vector_type(8)));

// ---- workspace layout (float offsets) ----
#define WS_XQ   0
#define WS_KC   (WS_XQ + BS*DH)
#define WS_VC   (WS_KC + BS*DH)
#define WS_RS   (WS_VC + BS*DH)
#define WS_SC   (WS_RS + BS*RREG)
#define WS_IN   (WS_SC + BS*RREG)
#define WS_AL   (WS_IN + BS*RREG)

// ---- output layout (float offsets, tuple order) ----
#define O_Y    0
#define O_PMK  (O_Y + BS*DH)
#define O_PMV  (O_PMK + BS*RREG*DH)
#define O_PMA  (O_PMV + BS*RREG*DH)
#define O_ELK  (O_PMA + BS*RREG)
#define O_ELV  (O_ELK + BS*RREG*DH)

__device__ __forceinline__ float wave_sum(float v) {
#pragma unroll
  for (int o = 16; o > 0; o >>= 1) v += __shfl_down(v, o, 32);
  return v;
}

// 256-thread (8-wave) block sum; sh must hold >= 8 floats
__device__ __forceinline__ float block_sum(float v, float* sh) {
  const int lane = threadIdx.x & 31;
  const int wid  = threadIdx.x >> 5;
  const int nw   = blockDim.x >> 5;
  v = wave_sum(v);
  if (lane == 0) sh[wid] = v;
  __syncthreads();
  if (wid == 0) {
    float x = (lane < nw) ? sh[lane] : 0.f;
    x = wave_sum(x);
    if (lane == 0) sh[0] = x;
  }
  __syncthreads();
  float r = sh[0];
  __syncthreads();
  return r;
}

// ------------------------------------------------------------------
// K1: Out[m][n] = sum_k In[m][k] * W[n][k] + bias[n]
// M=64, N=1024, K=1024, f32. One wave -> one 16x16 tile via
// V_WMMA_F32_16X16X4_F32, chaining C. grid=(16,4,2) block=128.
// A frag: lane(l) holds In[mb+l%16][k + 2*(l/16) + {0,1}]
// B frag: lane(l) holds W [nb+l%16][k + 2*(l/16) + {0,1}]  (W^T as 4x16)
// D layout: lane l, vgpr i -> M = mb + i + 8*(l/16), N = nb + l%16
// ------------------------------------------------------------------
__global__ void k1_gemm(const float* __restrict__ X, const float* __restrict__ H,
                        const float* __restrict__ Wk, const float* __restrict__ bk,
                        const float* __restrict__ Wv, const float* __restrict__ bv,
                        float* __restrict__ ws) {
  const int wave = threadIdx.x >> 5;
  const int lane = threadIdx.x & 31;
  const int half = lane >> 4;
  const int lm   = lane & 15;
  const int nb = (blockIdx.x * 4 + wave) * 16;
  const int mb = blockIdx.y * 16;
  const int which = blockIdx.z;

  const float* In   = which ? H  : X;
  const float* W    = which ? Wv : Wk;
  const float* bias = which ? bv : bk;
  float* Out = ws + (which ? WS_VC : WS_KC);

  const float* ar = In + (size_t)(mb + lm) * DH + 2 * half;
  const float* br = W  + (size_t)(nb + lm) * DH + 2 * half;

  v8f acc = {0.f, 0.f, 0.f, 0.f, 0.f, 0.f, 0.f, 0.f};
#if __has_builtin(__builtin_amdgcn_wmma_f32_16x16x4_f32)
  for (int k = 0; k < DH; k += 4) {
    v2f a = *(const v2f*)(ar + k);
    v2f b = *(const v2f*)(br + k);
    acc = __builtin_amdgcn_wmma_f32_16x16x4_f32(false, a, false, b,
                                                (short)0, acc, false, false);
  }
#else
  // scalar fallback matching the D layout
  for (int i = 0; i < 8; ++i) {
    const float* arow = In + (size_t)(mb + i + 8 * half) * DH;
    const float* brow = W  + (size_t)(nb + lm) * DH;
    float s = 0.f;
    for (int k = 0; k < DH; ++k) s += arow[k] * brow[k];
    acc[i] = s;
  }
#endif
  const float bb = bias[nb + lm];
#pragma unroll
  for (int i = 0; i < 8; ++i)
    Out[(size_t)(mb + i + 8 * half) * DH + nb + lm] = acc[i] + bb;
}

// ------------------------------------------------------------------
// K2: blocks 0..63   : x_q[b]  = normalize(x[b]);  y_pm[b] = 0
//     blocks 64..127 : k_cand  = normalize(k_pre)  (in place in ws)
// 256 threads, 4 floats/thread.
// ------------------------------------------------------------------
__global__ void k2_rows(const float* __restrict__ X, float* __restrict__ ws,
                        float* __restrict__ out) {
  __shared__ float sh[32];
  const int blk = blockIdx.x;
  const int d0  = threadIdx.x * 4;
  if (blk < BS) {
    const int b = blk;
    float4 xv = *(const float4*)(X + (size_t)b * DH + d0);
    float ss = xv.x * xv.x + xv.y * xv.y + xv.z * xv.z + xv.w * xv.w;
    ss = block_sum(ss, sh);
    const float inv = 1.f / (sqrtf(ss) + EPSV);
    *(float4*)(ws + WS_XQ + (size_t)b * DH + d0) =
        make_float4(xv.x * inv, xv.y * inv, xv.z * inv, xv.w * inv);
    *(float4*)(out + O_Y + (size_t)b * DH + d0) = make_float4(0.f, 0.f, 0.f, 0.f);
  } else {
    const int b = blk - BS;
    float* kp = ws + WS_KC + (size_t)b * DH;
    float4 xv = *(const float4*)(kp + d0);
    float ss = xv.x * xv.x + xv.y * xv.y + xv.z * xv.z + xv.w * xv.w;
    ss = block_sum(ss, sh);
    const float inv = 1.f / (sqrtf(ss) + EPSV);
    *(float4*)(kp + d0) = make_float4(xv.x * inv, xv.y * inv, xv.z * inv, xv.w * inv);
  }
}

// ------------------------------------------------------------------
// K3: per (b,r) row: read_scores, elig_K inv-norm, commit scores.
// grid=(512,64), 256 threads (4 floats/thread). Streams pm_K + elig_K.
// ------------------------------------------------------------------
__global__ void k3_scores(const float* __restrict__ pmK, const float* __restrict__ elK,
                          const float* __restrict__ pma, float* __restrict__ ws) {
  __shared__ float sh[32];
  const int r = blockIdx.x, b = blockIdx.y;
  const int d0 = threadIdx.x * 4;
  const size_t base = ((size_t)b * RREG + r) * DH + d0;
  float4 pk = *(const float4*)(pmK + base);
  float4 ek = *(const float4*)(elK + base);
  float4 xv = *(const float4*)(ws + WS_XQ + (size_t)b * DH + d0);
  float4 kv = *(const float4*)(ws + WS_KC + (size_t)b * DH + d0);
  const float e0 = RHO * ek.x + kv.x, e1 = RHO * ek.y + kv.y;
  const float e2 = RHO * ek.z + kv.z, e3 = RHO * ek.w + kv.w;
  float s1 = pk.x * xv.x + pk.y * xv.y + pk.z * xv.z + pk.w * xv.w;
  float nn = e0 * e0 + e1 * e1 + e2 * e2 + e3 * e3;
  float s2 = pk.x * e0 + pk.y * e1 + pk.z * e2 + pk.w * e3;
  s1 = block_sum(s1, sh);
  nn = block_sum(nn, sh);
  s2 = block_sum(s2, sh);
  if (threadIdx.x == 0) {
    const int idx = b * RREG + r;
    const float inv = 1.f / (sqrtf(nn) + EPSV);
    ws[WS_RS + idx] = s1;
    ws[WS_IN + idx] = inv;
    ws[WS_SC + idx] = s2 * inv - WEAKW * (pma[idx] * DECAYC);
  }
}

// ------------------------------------------------------------------
// K4: per batch b (512 threads, one per r): soft-top-8 softmax,
// alpha, decayed/clipped/budgeted pm_a.
// ------------------------------------------------------------------
__global__ void k4_topk(const float* __restrict__ pma,
                        const unsigned char* __restrict__ cmask,
                        float* __restrict__ ws, float* __restrict__ out) {
  __shared__ float work[RREG];
  __shared__ float rv[RREG];
  __shared__ int   ri[RREG];
  const int b = blockIdx.x, t = threadIdx.x;
  const float s = ws[WS_SC + b * RREG + t];
  work[t] = s;
  __syncthreads();
  float m0 = 0.f, thr = 0.f;
  for (int it = 0; it < TOPKC; ++it) {
    rv[t] = work[t];
    ri[t] = t;
    __syncthreads();
    for (int st = RREG / 2; st > 0; st >>= 1) {
      if (t < st) {
        const float v2 = rv[t + st];
        const int   i2 = ri[t + st];
        if (v2 > rv[t] || (v2 == rv[t] && i2 < ri[t])) { rv[t] = v2; ri[t] = i2; }
      }
      __syncthreads();
    }
    const float mv = rv[0];
    const int   mi = ri[0];
    if (it == 0) m0 = mv;
    thr = mv;
    if (t == mi) work[t] = -__builtin_inff();
    __syncthreads();
  }
  const float cm = cmask[b] ? 1.f : 0.f;
  const float e = (s >= thr) ? __expf((s - m0) / TAUC) : 0.f;
  rv[t] = e;
  __syncthreads();
  for (int st = RREG / 2; st > 0; st >>= 1) {
    if (t < st) rv[t] += rv[t + st];
    __syncthreads();
  }
  const float Z = rv[0];
  __syncthreads();
  const float alpha = (e / Z) * 0.5f * cm;
  const float a1 = pma[b * RREG + t] * DECAYC;
  const float a2 = a1 * (1.f - cm * (1.f - DECAYC));
  const float a3 = fminf(fmaxf(a2 + alpha, 0.f), A_MAXC);
  rv[t] = a3;
  __syncthreads();
  for (int st = RREG / 2; st > 0; st >>= 1) {
    if (t < st) rv[t] += rv[t + st];
    __syncthreads();
  }
  const float tot = rv[0];
  const float scale = fminf(1.f, BUDGETC / fmaxf(tot, EPSV));
  ws[WS_AL + b * RREG + t] = alpha;
  out[O_PMA + b * RREG + t] = a3 * scale;
}

// ------------------------------------------------------------------
// K5: fused update. grid=(64 r-groups, 64 b), 256 threads; each block
// handles 8 rows: blends+renormalizes pm_K/pm_V, writes elig outputs,
// accumulates y partials in registers, one atomicAdd burst at the end.
// ------------------------------------------------------------------
__global__ void k5_update(const float* __restrict__ pmK, const float* __restrict__ pmV,
                          const float* __restrict__ elK, const float* __restrict__ elV,
                          const float* __restrict__ pma,
                          const unsigned char* __restrict__ cmask,
                          float* __restrict__ ws, float* __restrict__ out) {
  __shared__ float sh[32];
  const int rg = blockIdx.x, b = blockIdx.y;
  const int d0 = threadIdx.x * 4;
  const float cm = cmask[b] ? 1.f : 0.f;
  const float keep = 1.f - cm;
  float4 kv = *(const float4*)(ws + WS_KC + (size_t)b * DH + d0);
  float4 vv = *(const float4*)(ws + WS_VC + (size_t)b * DH + d0);
  float y0 = 0.f, y1 = 0.f, y2 = 0.f, y3 = 0.f;

  for (int j = 0; j < 8; ++j) {
    const int r = rg * 8 + j;
    const int idx = b * RREG + r;
    const size_t base = ((size_t)idx) * DH + d0;
    if (j < 7) {  // prefetch next row's cachelines (global_prefetch_b8)
      __builtin_prefetch(pmK + base + DH, 0, 3);
      __builtin_prefetch(pmV + base + DH, 0, 3);
      __builtin_prefetch(elK + base + DH, 0, 3);
      __builtin_prefetch(elV + base + DH, 0, 3);
    }
    float4 pk = *(const float4*)(pmK + base);
    float4 pv = *(const float4*)(pmV + base);
    float4 ek = *(const float4*)(elK + base);
    float4 ev = *(const float4*)(elV + base);
    const float invn = ws[WS_IN + idx];
    const float a    = ws[WS_AL + idx];
    const float oma  = 1.f - a;

    const float eK0 = RHO * ek.x + kv.x, eK1 = RHO * ek.y + kv.y;
    const float eK2 = RHO * ek.z + kv.z, eK3 = RHO * ek.w + kv.w;
    const float eV0 = RHO * ev.x + vv.x, eV1 = RHO * ev.y + vv.y;
    const float eV2 = RHO * ev.z + vv.z, eV3 = RHO * ev.w + vv.w;

    const float bK0 = oma * pk.x + a * (eK0 * invn);
    const float bK1 = oma * pk.y + a * (eK1 * invn);
    const float bK2 = oma * pk.z + a * (eK2 * invn);
    const float bK3 = oma * pk.w + a * (eK3 * invn);
    const float bV0 = oma * pv.x + a * eV0;
    const float bV1 = oma * pv.y + a * eV1;
    const float bV2 = oma * pv.z + a * eV2;
    const float bV3 = oma * pv.w + a * eV3;

    float nK = bK0 * bK0 + bK1 * bK1 + bK2 * bK2 + bK3 * bK3;
    float nV = bV0 * bV0 + bV1 * bV1 + bV2 * bV2 + bV3 * bV3;
    nK = block_sum(nK, sh);
    nV = block_sum(nV, sh);
    const float sK = 1.f / (sqrtf(nK) + EPSV);
    const float sV = 1.f / (sqrtf(nV) + EPSV);

    *(float4*)(out + O_PMK + base) = make_float4(bK0 * sK, bK1 * sK, bK2 * sK, bK3 * sK);
    *(float4*)(out + O_PMV + base) = make_float4(bV0 * sV, bV1 * sV, bV2 * sV, bV3 * sV);
    *(float4*)(out + O_ELK + base) = make_float4(eK0 * keep, eK1 * keep, eK2 * keep, eK3 * keep);
    *(float4*)(out + O_ELV + base) = make_float4(eV0 * keep, eV1 * keep, eV2 * keep, eV3 * keep);

    const float wy = pma[idx] * ws[WS_RS + idx];  // original pm_a * read_score
    y0 += wy * pv.x; y1 += wy * pv.y; y2 += wy * pv.z; y3 += wy * pv.w;
  }
  float* yp = out + O_Y + (size_t)b * DH + d0;
  atomicAdd(yp + 0, y0);
  atomicAdd(yp + 1, y1);
  atomicAdd(yp + 2, y2);
  atomicAdd(yp + 3, y3);
}

extern "C" void kernel_launch(void* const* d_in, const int* in_sizes, int n_in,
                              void* d_out, int out_size, void* d_ws, size_t ws_size,
                              hipStream_t stream) {
  (void)in_sizes; (void)n_in; (void)out_size; (void)ws_size;
  const float* x   = (const float*)d_in[0];
  const float* h   = (const float*)d_in[1];
  const float* pmK = (const float*)d_in[2];
  const float* pmV = (const float*)d_in[3];
  const float* pma = (const float*)d_in[4];
  const float* elK = (const float*)d_in[5];
  const float* elV = (const float*)d_in[6];
  const float* Wk  = (const float*)d_in[7];
  const float* bk  = (const float*)d_in[8];
  const float* Wv  = (const float*)d_in[9];
  const float* bv  = (const float*)d_in[10];
  const unsigned char* cmask = (const unsigned char*)d_in[11];
  float* out = (float*)d_out;
  float* ws  = (float*)d_ws;

  k1_gemm<<<dim3(16, 4, 2), 128, 0, stream>>>(x, h, Wk, bk, Wv, bv, ws);
  k2_rows<<<dim3(2 * BS), 256, 0, stream>>>(x, ws, out);
  k3_scores<<<dim3(RREG, BS), 256, 0, stream>>>(pmK, elK, pma, ws);
  k4_topk<<<dim3(BS), RREG, 0, stream>>>(pma, cmask, ws, out);
  k5_update<<<dim3(RREG / 8, BS), 256, 0, stream>>>(pmK, pmV, elK, elV, pma, cmask, ws, out);
}